// ComputeLoss_15719580303700
// MI455X (gfx1250) — compile-verified
//
#include <hip/hip_runtime.h>

#define B_    16
#define N_    64
#define C_    80
#define H_    128
#define W_    128
#define RWIN  16
#define EPS32 1.1920929e-07f

typedef __attribute__((ext_vector_type(2))) float v2f;
typedef __attribute__((ext_vector_type(8))) float v8f;

// ---------------------------------------------------------------------------
// Wave-level f32 sum reduction via V_WMMA_F32_16X16X4_F32.
// Each lane puts its partial s into A (v0 = s, v1 = 0); B is all-ones.
// D[m][n] = sum_k A[m][k] = s_m + s_{m+16}, replicated across columns.
// Per lane, t = sum of the 8 D accumulator VGPRs:
//   lanes 0-15  -> sum of rows M=0..7  of D
//   lanes 16-31 -> sum of rows M=8..15 of D
// total = t(lane0) + t(lane16).  Exact f32 throughout.
// ---------------------------------------------------------------------------
__device__ __forceinline__ float wave_reduce_sum(float v) {
#if __has_builtin(__builtin_amdgcn_wmma_f32_16x16x4_f32)
    v2f a; a.x = v;    a.y = 0.0f;
    v2f b; b.x = 1.0f; b.y = 1.0f;
    v8f c = {};
    v8f d = __builtin_amdgcn_wmma_f32_16x16x4_f32(
        /*neg_a=*/false, a, /*neg_b=*/false, b,
        /*c_mod=*/(short)0, c, /*reuse_a=*/false, /*reuse_b=*/false);
    float t = ((d[0] + d[1]) + (d[2] + d[3])) + ((d[4] + d[5]) + (d[6] + d[7]));
    return __shfl(t, 0, 32) + __shfl(t, 16, 32);
#else
    for (int off = 16; off > 0; off >>= 1) v += __shfl_down(v, off, 32);
    return __shfl(v, 0, 32);
#endif
}

// Deterministic block reduction (fixed mapping, fixed order). Result valid on tid 0.
__device__ __forceinline__ float block_reduce_sum(float v, float* sm) {
    const int lane = threadIdx.x & 31;
    const int wid  = threadIdx.x >> 5;
    float w = wave_reduce_sum(v);          // EXEC all-ones here (uniform call site)
    if (lane == 0) sm[wid] = w;
    __syncthreads();
    float r = 0.0f;
    if (threadIdx.x == 0) {
        const int nw = (int)(blockDim.x >> 5);
        for (int i = 0; i < nw; ++i) r += sm[i];
    }
    __syncthreads();                       // allow sm reuse by a following call
    return r;
}

// ---------------------------------------------------------------------------
// Kernel 1: zero the ct map + 5 target maps (vectorized, grid-stride)
// ---------------------------------------------------------------------------
__global__ void zero_kernel(float4* __restrict__ p, long n4) {
    long i      = (long)blockIdx.x * blockDim.x + threadIdx.x;
    long stride = (long)gridDim.x * blockDim.x;
    float4 z = make_float4(0.f, 0.f, 0.f, 0.f);
    for (; i < n4; i += stride) p[i] = z;
}

// ---------------------------------------------------------------------------
// Kernel 2: scatter Gaussian windows with atomicMax (order-independent).
// One block per box; 256 threads cover the 33x33 window.
// ---------------------------------------------------------------------------
__global__ void scatter_kernel(const float* __restrict__ boxes,
                               const int*   __restrict__ labels,
                               float* __restrict__ ct,
                               float* __restrict__ wmap,
                               float* __restrict__ wh0, float* __restrict__ wh1,
                               float* __restrict__ of0, float* __restrict__ of1) {
    const int bi = blockIdx.x;            // 0 .. B*N-1
    const int b  = bi >> 6;               // N_ == 64
    const float x1 = boxes[bi * 4 + 0], y1 = boxes[bi * 4 + 1];
    const float x2 = boxes[bi * 4 + 2], y2 = boxes[bi * 4 + 3];
    const int   lab = labels[bi];

    const float wr = (float)W_ / 512.0f;
    const float hr = (float)H_ / 512.0f;
    const float cx = (x1 + x2) * wr * 0.5f;
    const float cy = (y1 + y2) * hr * 0.5f;
    const int   cxi = (int)floorf(cx);
    const int   cyi = (int)floorf(cy);
    const float sw = (x2 - x1) * wr;
    const float sh = (y2 - y1) * hr;

    // gaussian_radius(h=sh, w=sw, min_overlap=0.3)
    const float mo = 0.3f;
    const float hh = sh, ww = sw;
    const float b1 = hh + ww;
    const float c1 = ww * hh * (1.0f - mo) / (1.0f + mo);
    const float r1 = (b1 - sqrtf(b1 * b1 - 4.0f * c1)) * 0.5f;
    const float b2 = 2.0f * (hh + ww);
    const float c2 = (1.0f - mo) * ww * hh;
    const float r2 = (b2 - sqrtf(b2 * b2 - 16.0f * c2)) * 0.125f;
    const float a3 = 4.0f * mo;
    const float b3 = -2.0f * mo * (hh + ww);
    const float c3 = (mo - 1.0f) * ww * hh;
    const float r3 = (b3 + sqrtf(b3 * b3 - 4.0f * a3 * c3)) / (2.0f * a3);
    const float rf = fmaxf(0.0f, floorf(fminf(fminf(r1, r2), r3)));  // NaN -> 0
    const int   rad = (int)rf;
    const float dd = (float)(2 * rad + 1);
    const float sigma2 = (2.0f * dd / 6.0f) * (dd / 6.0f);
    const float inv_s2 = 1.0f / sigma2;

    const int win = 2 * RWIN + 1;  // 33
    for (int i = threadIdx.x; i < win * win; i += blockDim.x) {
        const int dy = i / win - RWIN;
        const int dx = i % win - RWIN;
        if (dx < -rad || dx > rad || dy < -rad || dy > rad) continue;
        const int yy = cyi + dy, xx = cxi + dx;
        if (yy < 0 || yy >= H_ || xx < 0 || xx >= W_) continue;
        const float dist2 = (float)(dx * dx + dy * dy);
        const float val = expf(-dist2 * inv_s2);
        if (val < EPS32) continue;        // reference zeroes sub-eps values
        const long idx = (((long)b * C_ + lab) * H_ + yy) * W_ + xx;
        // val >= 0, map init 0 -> int-bit atomicMax == float max
        atomicMax((int*)(ct + idx), __float_as_int(val));
    }
    if (threadIdx.x == 0) {
        const int p = (b * H_ + cyi) * W_ + cxi;
        wmap[p] = 1.0f;
        wh0[p]  = sw;
        wh1[p]  = sh;
        of0[p]  = cx - (float)cxi;
        of1[p]  = cy - (float)cyi;
    }
}

// ---------------------------------------------------------------------------
// Kernel 3: fused focal-loss reduction over B*C*H*W + positive count.
// ---------------------------------------------------------------------------
__global__ void center_loss_kernel(const float* __restrict__ pred,
                                   const float* __restrict__ ct,
                                   float* __restrict__ psum,
                                   float* __restrict__ pcnt, long n) {
    __shared__ float sm1[8];
    __shared__ float sm2[8];
    long i      = (long)blockIdx.x * blockDim.x + threadIdx.x;
    long stride = (long)gridDim.x * blockDim.x;
    float s = 0.0f, cnum = 0.0f;
    for (; i < n; i += stride) {
        __builtin_prefetch(&pred[i + stride], 0, 1);   // global_prefetch_b8
        const float p = pred[i];
        const float t = ct[i];
        const float omt  = 1.0f - t;
        const float negw = (omt * omt) * (omt * omt);
        float acc = -logf(1.0f - p + 1e-12f) * p * p * negw;
        if (t == 1.0f) {
            const float omp = 1.0f - p;
            acc += -logf(p + 1e-12f) * omp * omp;
            cnum += 1.0f;
        }
        s += acc;
    }
    const float bs = block_reduce_sum(s, sm1);
    const float bc = block_reduce_sum(cnum, sm2);
    if (threadIdx.x == 0) { psum[blockIdx.x] = bs; pcnt[blockIdx.x] = bc; }
}

// ---------------------------------------------------------------------------
// Kernel 4: sparse L1 losses (wh + offset) over B*H*W.
// ---------------------------------------------------------------------------
__global__ void reg_loss_kernel(const float* __restrict__ wh_pred,
                                const float* __restrict__ off_pred,
                                const float* __restrict__ wmap,
                                const float* __restrict__ wh0, const float* __restrict__ wh1,
                                const float* __restrict__ of0, const float* __restrict__ of1,
                                float* __restrict__ pwh, float* __restrict__ poff, int n) {
    __shared__ float sm1[8];
    __shared__ float sm2[8];
    int i      = blockIdx.x * blockDim.x + threadIdx.x;
    int stride = gridDim.x * blockDim.x;
    float swh = 0.0f, soff = 0.0f;
    const int HW = H_ * W_;
    for (; i < n; i += stride) {
        const float w = wmap[i];
        if (w != 0.0f) {
            const int b   = i / HW;
            const int rem = i - b * HW;
            const int i0  = (b * 2 + 0) * HW + rem;
            const int i1  = (b * 2 + 1) * HW + rem;
            swh  += (fabsf(wh_pred[i0]  - wh0[i]) + fabsf(wh_pred[i1]  - wh1[i])) * w;
            soff += (fabsf(off_pred[i0] - of0[i]) + fabsf(off_pred[i1] - of1[i])) * w;
        }
    }
    const float bs = block_reduce_sum(swh, sm1);
    const float bo = block_reduce_sum(soff, sm2);
    if (threadIdx.x == 0) { pwh[blockIdx.x] = bs; poff[blockIdx.x] = bo; }
}

// ---------------------------------------------------------------------------
// Kernel 5: deterministic final combine (single block, fixed-order sums).
// ---------------------------------------------------------------------------
__global__ void finalize_kernel(const float* __restrict__ psum,
                                const float* __restrict__ pcnt, int nc,
                                const float* __restrict__ pwh,
                                const float* __restrict__ poff, int nr,
                                float* __restrict__ out) {
    __shared__ float sm[8];
    float s = 0.0f, c = 0.0f, w = 0.0f, o = 0.0f;
    for (int i = threadIdx.x; i < nc; i += blockDim.x) { s += psum[i]; c += pcnt[i]; }
    for (int i = threadIdx.x; i < nr; i += blockDim.x) { w += pwh[i];  o += poff[i]; }
    const float S  = block_reduce_sum(s, sm);
    const float Cn = block_reduce_sum(c, sm);
    const float Ws = block_reduce_sum(w, sm);
    const float Os = block_reduce_sum(o, sm);
    if (threadIdx.x == 0) {
        const float af = fmaxf(1.0f, Cn);
        const float loss = S / (af + EPS32)
                         + 0.1f * Ws / (af * 2.0f + EPS32)
                         + 1.0f * Os / (af * 2.0f + EPS32);
        out[0] = loss;
    }
}

// ---------------------------------------------------------------------------
extern "C" void kernel_launch(void* const* d_in, const int* in_sizes, int n_in,
                              void* d_out, int out_size, void* d_ws, size_t ws_size,
                              hipStream_t stream) {
    (void)in_sizes; (void)n_in; (void)out_size; (void)ws_size;
    const float* center_pred = (const float*)d_in[0];  // B,C,H,W
    const float* wh_pred     = (const float*)d_in[1];  // B,2,H,W
    const float* off_pred    = (const float*)d_in[2];  // B,2,H,W
    const float* boxes       = (const float*)d_in[3];  // B,N,4
    const int*   labels      = (const int*)  d_in[4];  // B,N

    float* ws = (float*)d_ws;
    const long SZ_CT  = (long)B_ * C_ * H_ * W_;   // 20,971,520
    const long SZ_MAP = (long)B_ * H_ * W_;        //    262,144

    float* ct   = ws;
    float* wmap = ct   + SZ_CT;
    float* wh0  = wmap + SZ_MAP;
    float* wh1  = wh0  + SZ_MAP;
    float* of0  = wh1  + SZ_MAP;
    float* of1  = of0  + SZ_MAP;

    const int NBC = 2048;   // center-loss blocks
    const int NBR = 256;    // reg-loss blocks
    float* psum = of1  + SZ_MAP;
    float* pcnt = psum + NBC;
    float* pwh  = pcnt + NBC;
    float* poff = pwh  + NBR;

    const long zero_n = SZ_CT + 5 * SZ_MAP;        // multiple of 4
    zero_kernel<<<1024, 256, 0, stream>>>((float4*)ws, zero_n / 4);
    scatter_kernel<<<B_ * N_, 256, 0, stream>>>(boxes, labels, ct,
                                                wmap, wh0, wh1, of0, of1);
    center_loss_kernel<<<NBC, 256, 0, stream>>>(center_pred, ct, psum, pcnt, SZ_CT);
    reg_loss_kernel<<<NBR, 256, 0, stream>>>(wh_pred, off_pred, wmap,
                                             wh0, wh1, of0, of1, pwh, poff,
                                             (int)SZ_MAP);
    finalize_kernel<<<1, 256, 0, stream>>>(psum, pcnt, NBC, pwh, poff, NBR,
                                           (float*)d_out);
}